// NNModel_84396107366761
// MI455X (gfx1250) — compile-verified
//
#include <hip/hip_runtime.h>

// Problem constants (match reference)
#define HH    32          // hidden dim
#define NB    32          // batch
#define NPERB 512         // nodes per batch element
#define NNODE (NB*NPERB)  // 16384
#define EE    65536       // edges
#define NLAY  4
#define NCLS  16          // distinct edge_attr values
#define NL1   64
#define NL2   64

typedef float v2f __attribute__((ext_vector_type(2)));
typedef float v8f __attribute__((ext_vector_type(8)));

// ---- workspace layout (bytes) ----
static const size_t OFF_X0   = 0;                          // N*H f32  (2 MB)
static const size_t OFF_X1   = OFF_X0  + (size_t)NNODE*HH*4;
static const size_t OFF_AGG  = OFF_X1  + (size_t)NNODE*HH*4;
static const size_t OFF_WC   = OFF_AGG + (size_t)NNODE*HH*4;  // 16*32*32 f32 (64 KB)
static const size_t OFF_DEG  = OFF_WC  + (size_t)NCLS*HH*HH*4;
static const size_t OFF_INV  = OFF_DEG + (size_t)NNODE*4;
static const size_t OFF_CNT  = OFF_INV + (size_t)NNODE*4;     // ccount[16]+coffset[16]+cbase[17]+tbase[17]
static const size_t OFF_PERM = OFF_CNT + 1024;                // E ints (256 KB)

// ---------------- utility kernels ----------------
__global__ void k_zero(float* p, int n) {
  int i = blockIdx.x * blockDim.x + threadIdx.x;
  int stride = gridDim.x * blockDim.x;
  for (; i < n; i += stride) p[i] = 0.0f;
}

__global__ void k_embed(const int* __restrict__ x_nodes,
                        const float* __restrict__ node_emb,
                        float* __restrict__ x0) {
  int i = blockIdx.x * blockDim.x + threadIdx.x;
  if (i < NNODE * HH) {
    int n = i >> 5, h = i & 31;
    x0[i] = node_emb[x_nodes[n] * HH + h];
  }
}

__global__ void k_deg(const int* __restrict__ eidx, const int* __restrict__ eattr,
                      int* __restrict__ deg, int* __restrict__ ccount) {
  int e = blockIdx.x * blockDim.x + threadIdx.x;
  if (e < EE) {
    atomicAdd(&deg[eidx[EE + e]], 1);
    atomicAdd(&ccount[eattr[e]], 1);
  }
}

__global__ void k_scan(const int* __restrict__ ccount, int* __restrict__ cbase,
                       int* __restrict__ tbase) {
  if (threadIdx.x == 0 && blockIdx.x == 0) {
    int s = 0, t = 0;
    for (int c = 0; c < NCLS; c++) {
      cbase[c] = s; tbase[c] = t;
      s += ccount[c];
      t += (ccount[c] + 15) >> 4;
    }
    cbase[NCLS] = s; tbase[NCLS] = t;
  }
}

__global__ void k_invden(const int* __restrict__ deg, float* __restrict__ invden) {
  int n = blockIdx.x * blockDim.x + threadIdx.x;
  if (n < NNODE) invden[n] = 1.0f / fmaxf((float)deg[n], 1.0f);
}

__global__ void k_sort(const int* __restrict__ eattr, const int* __restrict__ cbase,
                       int* __restrict__ coffset, int* __restrict__ perm) {
  int e = blockIdx.x * blockDim.x + threadIdx.x;
  if (e < EE) {
    int c = eattr[e];
    int p = atomicAdd(&coffset[c], 1);
    perm[cbase[c] + p] = e;
  }
}

// Build the 16-entry W_c table for layer l:  W_c = relu(edge_emb[l][c]) @ lin_w[l] + lin_b[l]
__global__ void k_wc(const float* __restrict__ edge_emb_l,  // [NCLS][8]
                     const float* __restrict__ lin_w_l,     // [8][H*H]
                     const float* __restrict__ lin_b_l,     // [H*H]
                     float* __restrict__ Wc) {              // [NCLS][H][H]
  int c = blockIdx.x;
  __shared__ float e8[8];
  if (threadIdx.x < 8) {
    float v = edge_emb_l[c * 8 + threadIdx.x];
    e8[threadIdx.x] = v > 0.0f ? v : 0.0f;
  }
  __syncthreads();
  for (int idx = threadIdx.x; idx < HH * HH; idx += blockDim.x) {
    float s = lin_b_l[idx];
#pragma unroll
    for (int j = 0; j < 8; j++) s += e8[j] * lin_w_l[j * HH * HH + idx];
    Wc[c * HH * HH + idx] = s;
  }
}

// ---------------- WMMA message kernel ----------------
// One wave per tile of 16 same-class edges:
//   msg[16x32] = A(x[src], 16x32) @ W_c(32x32)   via 2 col-tiles x 8 k-steps of
//   v_wmma_f32_16x16x4_f32, then atomic scatter into agg[dst].
__global__ __launch_bounds__(32) void k_msg(const int* __restrict__ perm,
                                            const int* __restrict__ eidx,
                                            const float* __restrict__ x_in,
                                            const float* __restrict__ Wc,
                                            float* __restrict__ agg,
                                            const int* __restrict__ cbase,
                                            const int* __restrict__ tbase) {
  int t = blockIdx.x;
  if (t >= tbase[NCLS]) return;
  int c = 0;
#pragma unroll
  for (int i = 0; i < NCLS - 1; i++)
    if (t >= tbase[i + 1]) c = i + 1;
  int pos0  = cbase[c] + (t - tbase[c]) * 16;
  int nrows = cbase[c + 1] - pos0;
  if (nrows > 16) nrows = 16;

  __shared__ float Wl[HH][HH];
  __shared__ float A[16][HH + 1];
  __shared__ int dstr[16];
  __shared__ int srcr[16];

  int lane = threadIdx.x;
#pragma unroll
  for (int i = 0; i < HH; i++) Wl[i][lane] = Wc[c * HH * HH + i * HH + lane];
  if (lane < 16) {
    if (lane < nrows) {
      int e = perm[pos0 + lane];
      srcr[lane] = eidx[e];        // src
      dstr[lane] = eidx[EE + e];   // dst
    } else {
      srcr[lane] = 0;
      dstr[lane] = -1;
    }
  }
  __syncthreads();
#pragma unroll
  for (int i = 0; i < 16; i++)
    A[i][lane] = (i < nrows) ? x_in[srcr[i] * HH + lane] : 0.0f;
  __syncthreads();

  int half = lane >> 4, l16 = lane & 15;
  v8f acc0 = {}; v8f acc1 = {};
#pragma unroll
  for (int kk = 0; kk < HH; kk += 4) {
    v2f a, b0, b1;
    // A 16x4 layout: lanes 0-15 hold K=kk..kk+1, lanes 16-31 hold K=kk+2..kk+3 (row = lane&15)
    a[0] = A[l16][kk + 2 * half];
    a[1] = A[l16][kk + 2 * half + 1];
    // B 4x16 layout (mirror): lanes 0-15 hold K rows kk..kk+1, lanes 16-31 kk+2..kk+3 (col = lane&15)
    b0[0] = Wl[kk + 2 * half][l16];
    b0[1] = Wl[kk + 2 * half + 1][l16];
    b1[0] = Wl[kk + 2 * half][16 + l16];
    b1[1] = Wl[kk + 2 * half + 1][16 + l16];
    acc0 = __builtin_amdgcn_wmma_f32_16x16x4_f32(false, a, false, b0, (short)0, acc0, false, false);
    acc1 = __builtin_amdgcn_wmma_f32_16x16x4_f32(false, a, false, b1, (short)0, acc1, false, false);
  }
  // C layout: VGPR v -> row v (lanes 0-15) / row v+8 (lanes 16-31), col = lane&15 (+16 for tile 1)
#pragma unroll
  for (int v = 0; v < 8; v++) {
    int row = v + 8 * half;
    int d = dstr[row];
    if (d >= 0) {
      atomicAdd(&agg[d * HH + l16],      acc0[v]);
      atomicAdd(&agg[d * HH + 16 + l16], acc1[v]);
    }
  }
}

// ---------------- WMMA root-update kernel ----------------
// x_out = relu?( agg*invden + x_in @ root_w + conv_b ), and clears agg for next layer.
__global__ __launch_bounds__(32) void k_update(const float* __restrict__ x_in,
                                               const float* __restrict__ rootw,
                                               const float* __restrict__ convb,
                                               float* __restrict__ agg,
                                               const float* __restrict__ invden,
                                               float* __restrict__ x_out,
                                               int do_relu) {
  int m0 = blockIdx.x * 16;
  __shared__ float Wl[HH][HH];
  __shared__ float A[16][HH + 1];
  int lane = threadIdx.x;
#pragma unroll
  for (int i = 0; i < HH; i++) Wl[i][lane] = rootw[i * HH + lane];
#pragma unroll
  for (int i = 0; i < 16; i++) A[i][lane] = x_in[(m0 + i) * HH + lane];
  __syncthreads();

  int half = lane >> 4, l16 = lane & 15;
  v8f acc0 = {}; v8f acc1 = {};
#pragma unroll
  for (int kk = 0; kk < HH; kk += 4) {
    v2f a, b0, b1;
    a[0]  = A[l16][kk + 2 * half];
    a[1]  = A[l16][kk + 2 * half + 1];
    b0[0] = Wl[kk + 2 * half][l16];
    b0[1] = Wl[kk + 2 * half + 1][l16];
    b1[0] = Wl[kk + 2 * half][16 + l16];
    b1[1] = Wl[kk + 2 * half + 1][16 + l16];
    acc0 = __builtin_amdgcn_wmma_f32_16x16x4_f32(false, a, false, b0, (short)0, acc0, false, false);
    acc1 = __builtin_amdgcn_wmma_f32_16x16x4_f32(false, a, false, b1, (short)0, acc1, false, false);
  }
#pragma unroll
  for (int v = 0; v < 8; v++) {
    int m = m0 + v + 8 * half;
    float inv = invden[m];
    int i0 = m * HH + l16;
    int i1 = m * HH + 16 + l16;
    float r0 = acc0[v] + agg[i0] * inv + convb[l16];
    float r1 = acc1[v] + agg[i1] * inv + convb[16 + l16];
    if (do_relu) { r0 = fmaxf(r0, 0.0f); r1 = fmaxf(r1, 0.0f); }
    x_out[i0] = r0; x_out[i1] = r1;
    agg[i0] = 0.0f; agg[i1] = 0.0f;   // ready for next layer
  }
}

// ---------------- readout ----------------
__global__ void k_readout(const float* __restrict__ x, const int* __restrict__ metal_idx,
                          const int* __restrict__ loop_edge, const int* __restrict__ loop_pair,
                          const float* __restrict__ f_w, const float* __restrict__ f_b,
                          float* __restrict__ out) {
  int b = blockIdx.x;
  int j = threadIdx.x;
  const float* xb = x + (size_t)b * NPERB * HH;
  const float* xm = xb + metal_idx[b] * HH;
  float fh = f_b[0];
#pragma unroll
  for (int h = 0; h < HH; h++) fh += xm[h] * f_w[h];
  int s, t;
  if (j < NL1) {
    s = loop_edge[(b * NL1 + j) * 2 + 0];
    t = loop_edge[(b * NL1 + j) * 2 + 1];
  } else {
    int jj = j - NL1;
    s = loop_pair[(b * NL2 + jj) * 2 + 0];
    t = loop_pair[(b * NL2 + jj) * 2 + 1];
  }
  const float* xs = xb + s * HH;
  const float* xt = xb + t * HH;
  float v = 0.0f;
#pragma unroll
  for (int h = 0; h < HH; h++) v += xm[h] * (xs[h] + xt[h]) - xs[h] * xt[h];
  out[b * (NL1 + NL2) + j] = (j < NL1) ? (v - fh) : (-v);
}

// ---------------- launcher ----------------
extern "C" void kernel_launch(void* const* d_in, const int* in_sizes, int n_in,
                              void* d_out, int out_size, void* d_ws, size_t ws_size,
                              hipStream_t stream) {
  (void)in_sizes; (void)n_in; (void)out_size; (void)ws_size;
  const int*   x_nodes    = (const int*)  d_in[0];
  const int*   edge_index = (const int*)  d_in[1];
  const int*   edge_attr  = (const int*)  d_in[2];
  const int*   metal_idx  = (const int*)  d_in[3];
  const int*   loop_edge  = (const int*)  d_in[4];
  const int*   loop_pair  = (const int*)  d_in[5];
  const float* node_emb   = (const float*)d_in[6];
  const float* edge_emb   = (const float*)d_in[7];
  const float* lin_w      = (const float*)d_in[8];
  const float* lin_b      = (const float*)d_in[9];
  const float* root_w     = (const float*)d_in[10];
  const float* conv_b     = (const float*)d_in[11];
  const float* f_w        = (const float*)d_in[12];
  const float* f_b        = (const float*)d_in[13];

  char* ws = (char*)d_ws;
  float* x0     = (float*)(ws + OFF_X0);
  float* x1     = (float*)(ws + OFF_X1);
  float* agg    = (float*)(ws + OFF_AGG);
  float* Wc     = (float*)(ws + OFF_WC);
  int*   deg    = (int*)  (ws + OFF_DEG);
  float* invden = (float*)(ws + OFF_INV);
  int*   ccount = (int*)  (ws + OFF_CNT);
  int*   coffset= ccount + NCLS;
  int*   cbase  = coffset + NCLS;      // 17 entries
  int*   tbase  = cbase + NCLS + 1;    // 17 entries
  int*   perm   = (int*)  (ws + OFF_PERM);

  // init / zero (ws is poisoned; agg must be zero, counters must be zero)
  k_zero<<<256, 256, 0, stream>>>(agg, NNODE * HH);
  k_zero<<<64, 256, 0, stream>>>((float*)deg, NNODE);
  k_zero<<<1, 64, 0, stream>>>((float*)ccount, 2 * NCLS);

  k_embed<<<(NNODE * HH) / 256, 256, 0, stream>>>(x_nodes, node_emb, x0);
  k_deg<<<EE / 256, 256, 0, stream>>>(edge_index, edge_attr, deg, ccount);
  k_scan<<<1, 32, 0, stream>>>(ccount, cbase, tbase);
  k_invden<<<NNODE / 256, 256, 0, stream>>>(deg, invden);
  k_sort<<<EE / 256, 256, 0, stream>>>(edge_attr, cbase, coffset, perm);

  float* xin = x0;
  float* xout = x1;
  for (int l = 0; l < NLAY; l++) {
    k_wc<<<NCLS, 256, 0, stream>>>(edge_emb + (size_t)l * NCLS * 8,
                                   lin_w + (size_t)l * 8 * HH * HH,
                                   lin_b + (size_t)l * HH * HH, Wc);
    k_msg<<<EE / 16 + NCLS, 32, 0, stream>>>(perm, edge_index, xin, Wc, agg, cbase, tbase);
    k_update<<<NNODE / 16, 32, 0, stream>>>(xin, root_w + (size_t)l * HH * HH,
                                            conv_b + (size_t)l * HH, agg, invden, xout,
                                            (l < NLAY - 1) ? 1 : 0);
    float* tmp = xin; xin = xout; xout = tmp;
  }

  k_readout<<<NB, NL1 + NL2, 0, stream>>>(xin, metal_idx, loop_edge, loop_pair, f_w, f_b,
                                          (float*)d_out);
}